// CTC_CRF_LOSS_16312285790337
// MI455X (gfx1250) — compile-verified
//
#include <hip/hip_runtime.h>
#include <hip/hip_bf16.h>

// Problem constants (from reference setup_inputs)
#define B_ 16
#define T_ 800
#define V_ 4000
#define L_ 100
#define S_ 201          // 2L+1
#define NEGF (-1.0e30f)
#define LAMB_ 0.1f

typedef __attribute__((ext_vector_type(2))) float v2f;
typedef __attribute__((ext_vector_type(8))) float v8f;

__device__ __forceinline__ float lae(float a, float b) {
  // jnp.logaddexp semantics for finite / -1e30 surrogate values
  float mx = fmaxf(a, b);
  float mn = fminf(a, b);
  return mx + log1pf(expf(mn - mx));
}

// ---------------------------------------------------------------------------
// Phase 1: per-(b,t) logsumexp over V (streaming, memory-bound) + emit gather.
// One workgroup (256 thr) handles 16 consecutive rows (same b, T%16==0).
// Loads are strided float2 (global_load_b64): 16 lanes x 8B = 128B contiguous
// per row-half; V=4000 = 16 threads * 125 float2 iters exactly.
// Final 4-partial-per-row sum across all 16 rows is one exact-f32
// V_WMMA_F32_16X16X4_F32:  D = A(16x4 partials) x ones(4x16).
// ---------------------------------------------------------------------------
__global__ __launch_bounds__(256) void p1_lse_emit(const float* __restrict__ logits,
                                                   const int* __restrict__ labels,
                                                   float* __restrict__ lse,
                                                   float* __restrict__ emit) {
  __shared__ float red_m[16][17];
  __shared__ float red_s[16][17];
  __shared__ float lse_s[16];

  const int tid  = threadIdx.x;
  const int r    = tid >> 4;   // row within tile [0,16)
  const int c    = tid & 15;   // partial index within row [0,16)
  const int row0 = blockIdx.x << 4;
  const int b    = row0 / T_;

  // Online (max-rescaled) exp-sum; vectorized b64 streaming loads.
  const float2* rowp2 =
      (const float2*)(logits + (size_t)(row0 + r) * V_);
  float m = -3.0e38f, s = 0.0f;
#pragma unroll 4
  for (int k = 0; k < V_ / 32; ++k) {           // 125 iterations
    float2 x2 = rowp2[c + 16 * k];
    float x = x2.x;
    if (x > m) { s = s * expf(m - x) + 1.0f; m = x; }
    else       { s += expf(x - m); }
    x = x2.y;
    if (x > m) { s = s * expf(m - x) + 1.0f; m = x; }
    else       { s += expf(x - m); }
  }

  // Row max across 16 partials.
  red_m[r][c] = m;
  __syncthreads();
  for (int st = 8; st > 0; st >>= 1) {
    if (c < st) red_m[r][c] = fmaxf(red_m[r][c], red_m[r][c + st]);
    __syncthreads();
  }
  const float M = red_m[r][0];

  // Rescale partial sums to common max, reduce 16 -> 4 partials per row.
  red_s[r][c] = s * expf(m - M);
  __syncthreads();
  if (c < 8) red_s[r][c] += red_s[r][c + 8];
  __syncthreads();
  if (c < 4) red_s[r][c] += red_s[r][c + 4];
  __syncthreads();

  // Exact f32 WMMA row-sum: A[m][k] = partial k of row m, B = ones.
  // A 16x4 layout: lanes 0-15 hold (K=0,K=1) in (v0,v1); lanes 16-31 hold (K=2,K=3).
  // Executed by ALL waves (full EXEC, as WMMA requires); results identical.
  const int lane = tid & 31;
  v2f a;
  if (lane < 16) { a.x = red_s[lane][0];      a.y = red_s[lane][1]; }
  else           { a.x = red_s[lane - 16][2]; a.y = red_s[lane - 16][3]; }
  v2f ones; ones.x = 1.0f; ones.y = 1.0f;
  v8f cz = {};
  v8f d = __builtin_amdgcn_wmma_f32_16x16x4_f32(false, a, false, ones,
                                                (short)0, cz, false, false);
  __syncthreads();
  // C/D layout: VGPR i, lane 0 -> M=i ; lane 16 -> M=i+8 (value broadcast over N).
  if (tid == 0) {
#pragma unroll
    for (int i = 0; i < 8; ++i) lse_s[i] = red_m[i][0] + logf(d[i]);
  } else if (tid == 16) {
#pragma unroll
    for (int i = 0; i < 8; ++i) lse_s[8 + i] = red_m[8 + i][0] + logf(d[i]);
  }
  __syncthreads();
  if (c == 0) lse[row0 + r] = lse_s[r];

  // Emit gather: emit[b,t,s] = logits[b,t,ext(s)] - lse[b,t]  (rows still hot in cache)
  const int* lab = labels + b * L_;
  for (int i = tid; i < 16 * S_; i += 256) {
    int rr = i / S_;
    int ss = i - rr * S_;
    int ext = (ss & 1) ? lab[ss >> 1] : 0;
    float e = logits[(size_t)(row0 + rr) * V_ + ext] - lse_s[rr];
    emit[(size_t)(row0 + rr) * S_ + ss] = e;
  }
}

// ---------------------------------------------------------------------------
// Phase 2: CTC forward DP (one block per batch element, thread s = state s),
// double-buffered alpha in LDS; emit[t+1] load software-pipelined so global
// latency hides under the barrier + logaddexp chain. Plus CRF denominator.
// ---------------------------------------------------------------------------
__global__ __launch_bounds__(256) void p2_dp(const float* __restrict__ emit,
                                             const float* __restrict__ lse,
                                             const int* __restrict__ labels,
                                             const int* __restrict__ in_len,
                                             const int* __restrict__ lab_len,
                                             float* __restrict__ cost) {
  __shared__ float bufA[S_ + 8];
  __shared__ float bufB[S_ + 8];
  __shared__ float red[256];

  const int b = blockIdx.x;
  const int s = threadIdx.x;
  const int* lab = labels + b * L_;
  const float* em = emit + (size_t)b * T_ * S_;
  const int myT = in_len[b];
  const bool act = (s < S_);

  bool allow = false;
  float ecur = 0.0f;
  if (act) {
    if ((s & 1) && s >= 2) {
      int ext   = lab[s >> 1];
      int extm2 = lab[(s - 2) >> 1];
      allow = (ext != extm2);   // skip transition s-2 -> s
    }
    float a0 = (s == 0) ? em[0] : (s == 1) ? em[1] : NEGF;
    bufA[s] = a0;
    ecur = em[(size_t)S_ + s];  // emit for t=1, prefetched
  }
  __syncthreads();

  for (int t = 1; t < T_; ++t) {
    const float* rd = (t & 1) ? bufA : bufB;
    float*       wr = (t & 1) ? bufB : bufA;
    float enext = 0.0f;
    if (act && (t + 1 < T_)) enext = em[(size_t)(t + 1) * S_ + s];  // issue early
    if (act) {
      float as = rd[s];
      float a1 = (s >= 1) ? rd[s - 1] : NEGF;
      float a2 = allow    ? rd[s - 2] : NEGF;
      float v  = lae(lae(as, a1), a2) + ecur;
      if (t >= myT) v = as;     // freeze finished sequences
      wr[s] = v;
    }
    ecur = enext;
    __syncthreads();
  }
  const float* fin = ((T_ - 1) & 1) ? bufB : bufA;

  // CRF denominator: sum_{t < myT} lse[b,t]
  float local = 0.0f;
  const float* lrow = lse + b * T_;
  for (int t = s; t < myT; t += 256) local += lrow[t];
  red[s] = local;
  __syncthreads();
  for (int st = 128; st > 0; st >>= 1) {
    if (s < st) red[s] += red[s + st];
    __syncthreads();
  }

  if (s == 0) {
    int end   = 2 * lab_len[b];                 // >= 2 since label_len >= 1
    float nll = -lae(fin[end], fin[end - 1]);
    cost[b]   = red[0] - (1.0f + LAMB_) * nll;  // costs_den - 1.1 * costs_ctc
  }
}

// ---------------------------------------------------------------------------
// Phase 3: mean over batch.
// ---------------------------------------------------------------------------
__global__ void p3_final(const float* __restrict__ cost, float* __restrict__ out) {
  if (threadIdx.x == 0) {
    float sum = 0.0f;
    for (int b = 0; b < B_; ++b) sum += cost[b];
    out[0] = sum / (float)B_;
  }
}

extern "C" void kernel_launch(void* const* d_in, const int* in_sizes, int n_in,
                              void* d_out, int out_size, void* d_ws, size_t ws_size,
                              hipStream_t stream) {
  const float* logits  = (const float*)d_in[0];
  const int*   labels  = (const int*)d_in[1];
  const int*   in_len  = (const int*)d_in[2];
  const int*   lab_len = (const int*)d_in[3];

  float* ws   = (float*)d_ws;
  float* lse  = ws;                                  // B*T floats
  float* emit = ws + (size_t)B_ * T_;                // B*T*S floats (~10.3 MB)
  float* cost = emit + (size_t)B_ * T_ * S_;         // B floats

  p1_lse_emit<<<(B_ * T_) / 16, 256, 0, stream>>>(logits, labels, lse, emit);
  p2_dp<<<B_, 256, 0, stream>>>(emit, lse, labels, in_len, lab_len, cost);
  p3_final<<<1, 64, 0, stream>>>(cost, (float*)d_out);
}